// MIDGCN_87058987090446
// MI455X (gfx1250) — compile-verified
//
#include <hip/hip_runtime.h>
#include <hip/hip_bf16.h>

#define B_   16
#define C_   2048
#define T_   128
#define GIN_ 192
#define OUT_ 96
#define NEGV (-1.0e9f)

typedef __attribute__((ext_vector_type(2))) float v2f;
typedef __attribute__((ext_vector_type(8))) float v8f;

#if defined(__has_builtin)
#if __has_builtin(__builtin_amdgcn_global_load_async_to_lds_b128) && \
    __has_builtin(__builtin_amdgcn_s_wait_asynccnt)
#define HAVE_ASYNC_LDS 1
#endif
#endif

__device__ __forceinline__ v8f wmma4(v2f a, v2f b, v8f c) {
    // D = A(16x4) * B(4x16) + C, full fp32 (V_WMMA_F32_16X16X4_F32)
    return __builtin_amdgcn_wmma_f32_16x16x4_f32(
        /*neg_a=*/false, a, /*neg_b=*/false, b,
        /*c_mod=*/(short)0, c, /*reuse_a=*/false, /*reuse_b=*/false);
}

#if HAVE_ASYNC_LDS
// Builtin expects (AS(1) int4* src, AS(3) int4* dst, imm offset, imm cpol)
typedef __attribute__((vector_size(16))) int v4i;
typedef __attribute__((address_space(1))) v4i glb_v4i;
typedef __attribute__((address_space(3))) v4i lds_v4i;
// Copy one contiguous 16x192 f32 tile (12288 B) global -> LDS asynchronously.
// 768 16-byte chunks / 32 lanes = 24 async b128 ops (ASYNCcnt += 24).
__device__ __forceinline__ void async_tile(const float* __restrict__ src,
                                           float* __restrict__ dst, int lane) {
#pragma unroll
    for (int i = 0; i < 24; ++i) {
        const int chunk = i * 32 + lane;
        __builtin_amdgcn_global_load_async_to_lds_b128(
            (glb_v4i*)(src + chunk * 4),
            (lds_v4i*)(dst + chunk * 4),
            /*offset=*/0, /*cpol=*/0);
    }
}
#endif

// ---------------------------------------------------------------------------
// Generic 16x16-tile WMMA fp32 GEMM, one wave per output tile.
//   C[b, m, n] = act( sum_k A[b,m,k]*B[b,k,n] + bias[n] + R[b,m,n] )
// Strides let us express transposed operands (e.g. agg^T) without copies.
// ACT: 0=none, 1=relu, 2=tanh
// ---------------------------------------------------------------------------
template <int ACT, bool BIAS, bool RES>
__global__ __launch_bounds__(32)
void gemm16(const float* __restrict__ A, long sAm, long sAk, long sAb,
            const float* __restrict__ Bm, long sBk, long sBn, long sBb,
            const float* __restrict__ bias,
            const float* __restrict__ R, long sRm, long sRn, long sRb,
            float* __restrict__ Cp, long sCm, long sCn, long sCb,
            int K)
{
    const int m0   = blockIdx.x * 16;
    const int n0   = blockIdx.y * 16;
    const int b    = blockIdx.z;
    const int lane = threadIdx.x;
    const int lo   = lane & 15;   // A: row M, B: col N, C/D: col N
    const int hi   = lane >> 4;   // selects K pair (A/B) / M half (C/D)

    const float* Ab = A  + (long)b * sAb + (long)(m0 + lo) * sAm;
    const float* Bb = Bm + (long)b * sBb + (long)(n0 + lo) * sBn;

    v8f acc = {};
#pragma unroll 4
    for (int k = 0; k < K; k += 4) {
        const long ka = (long)(k + 2 * hi);
        v2f av, bv;
        av.x = Ab[ ka      * sAk];
        av.y = Ab[(ka + 1) * sAk];
        bv.x = Bb[ ka      * sBk];
        bv.y = Bb[(ka + 1) * sBk];
        acc = wmma4(av, bv, acc);
    }

    const float bn = BIAS ? bias[n0 + lo] : 0.0f;
#pragma unroll
    for (int r = 0; r < 8; ++r) {
        const int m = m0 + r + 8 * hi;
        const int n = n0 + lo;
        float v = acc[r] + bn;
        if (RES) v += R[(long)b * sRb + (long)m * sRm + (long)n * sRn];
        if (ACT == 1) v = fmaxf(v, 0.0f);
        if (ACT == 2) v = tanhf(v);
        Cp[(long)b * sCb + (long)m * sCm + (long)n * sCn] = v;
    }
}

// ---------------------------------------------------------------------------
// Per-row pipeline (VALU): dyn -> id_embeds -> probs -> cluster_embeds ->
// graph_emb. One lane per (b,c) row. Emits G (B,C,32) and V=gcn_input
// (B,C,192) = [xs | id_embeds | cluster_embeds].
// ---------------------------------------------------------------------------
__global__ __launch_bounds__(256)
void row_pipeline(const float* __restrict__ x,       // (B*C,128)
                  const float* __restrict__ H,       // (B*C,64) = relu(x@w1+b1)
                  const float* __restrict__ id_emb,  // (C,32)
                  const float* __restrict__ w2,      // (64,32)
                  const float* __restrict__ b2,      // (32)
                  const float* __restrict__ ce,      // (16,32) cluster_emb
                  const float* __restrict__ wic,     // (32,32)
                  const float* __restrict__ bic,     // (32)
                  const float* __restrict__ wg,      // (64,32)
                  const float* __restrict__ bg,      // (32)
                  float* __restrict__ Gout,          // (B*C,32)
                  float* __restrict__ Vout)          // (B*C,192)
{
    const int row = blockIdx.x * blockDim.x + threadIdx.x;
    if (row >= B_ * C_) return;
    const int c = row & (C_ - 1);

    // dyn = H @ w2 + b2
    float dyn[32];
#pragma unroll
    for (int j = 0; j < 32; ++j) dyn[j] = b2[j];
    const float* h = H + (long)row * 64;
    for (int i = 0; i < 64; ++i) {
        const float hv = h[i];
#pragma unroll
        for (int j = 0; j < 32; ++j) dyn[j] = fmaf(hv, w2[i * 32 + j], dyn[j]);
    }

    // id_embeds
    float ide[32];
#pragma unroll
    for (int j = 0; j < 32; ++j) ide[j] = id_emb[c * 32 + j] + dyn[j];

    // id_for_clustering = ide @ wic + bic
    float ic[32];
#pragma unroll
    for (int j = 0; j < 32; ++j) ic[j] = bic[j];
    for (int l = 0; l < 32; ++l) {
        const float iv = ide[l];
#pragma unroll
        for (int j = 0; j < 32; ++j) ic[j] = fmaf(iv, wic[l * 32 + j], ic[j]);
    }

    // probs = softmax(ic . cluster_emb_n), cluster_embeds = probs @ ce
    float lg[16];
    float mx = -3.0e38f;
#pragma unroll
    for (int n = 0; n < 16; ++n) {
        float s = 0.0f;
#pragma unroll
        for (int l = 0; l < 32; ++l) s = fmaf(ic[l], ce[n * 32 + l], s);
        lg[n] = s;
        mx = fmaxf(mx, s);
    }
    float den = 0.0f;
#pragma unroll
    for (int n = 0; n < 16; ++n) { lg[n] = __expf(lg[n] - mx); den += lg[n]; }
    const float inv = 1.0f / den;
    float cl[32];
#pragma unroll
    for (int j = 0; j < 32; ++j) cl[j] = 0.0f;
    for (int n = 0; n < 16; ++n) {
        const float p = lg[n] * inv;
#pragma unroll
        for (int j = 0; j < 32; ++j) cl[j] = fmaf(p, ce[n * 32 + j], cl[j]);
    }

    // graph_emb = [ide|cl] @ wg + bg
    float ge[32];
#pragma unroll
    for (int g = 0; g < 32; ++g) ge[g] = bg[g];
    for (int l = 0; l < 32; ++l) {
        const float iv = ide[l];
#pragma unroll
        for (int g = 0; g < 32; ++g) ge[g] = fmaf(iv, wg[l * 32 + g], ge[g]);
    }
    for (int l = 0; l < 32; ++l) {
        const float cv = cl[l];
#pragma unroll
        for (int g = 0; g < 32; ++g) ge[g] = fmaf(cv, wg[(32 + l) * 32 + g], ge[g]);
    }

    float* grow = Gout + (long)row * 32;
#pragma unroll
    for (int g = 0; g < 32; ++g) grow[g] = ge[g];

    float* vrow = Vout + (long)row * GIN_;
    const float* xrow = x + (long)row * T_;
#pragma unroll 4
    for (int t = 0; t < T_; ++t) vrow[t] = xrow[t];
#pragma unroll
    for (int j = 0; j < 32; ++j) { vrow[128 + j] = ide[j]; vrow[160 + j] = cl[j]; }
}

// ---------------------------------------------------------------------------
// Fused attention: adj = softmax(mask(relu(G G^T))), AGG = adj @ V.
// One wave handles 16 output rows of one batch; streams m in 16-tiles with
// online softmax (flash style) so the 2048x2048 adj never hits memory.
// V tiles are staged into LDS with double-buffered async-to-LDS DMA
// (ASYNCcnt / s_wait_asynccnt) so the fetch overlaps the S-tile WMMAs.
// Row statistics live in 16-lane groups (D-layout rows), reduced by shfl_xor.
// ---------------------------------------------------------------------------
__global__ __launch_bounds__(32)
void flash_adj_agg(const float* __restrict__ G,   // (B,C,32)
                   const float* __restrict__ V,   // (B,C,192)
                   float* __restrict__ AGG)       // (B,C,192)
{
    const int b    = blockIdx.z;
    const int o0   = blockIdx.x * 16;
    const int lane = threadIdx.x;
    const int lo   = lane & 15;
    const int hi   = lane >> 4;

    const float* Gb = G + (long)b * C_ * 32;
    const float* Vb = V + (long)b * C_ * GIN_;

    // Q tiles in A-layout (M=o rows, K=feature), preloaded for all 8 K-steps
    v2f q[8];
    {
        const float* qrow = Gb + (long)(o0 + lo) * 32 + 2 * hi;
#pragma unroll
        for (int kt = 0; kt < 8; ++kt) { q[kt].x = qrow[4 * kt]; q[kt].y = qrow[4 * kt + 1]; }
    }

    float mrun[8], lrun[8];
    v8f acc[12];
#pragma unroll
    for (int r = 0; r < 8; ++r) { mrun[r] = -3.0e38f; lrun[r] = 0.0f; }
#pragma unroll
    for (int nt = 0; nt < 12; ++nt) acc[nt] = (v8f){};

    __shared__ float pshr[16 * 16];   // P tile D-layout -> A-layout via LDS
#if HAVE_ASYNC_LDS
    __shared__ __align__(16) float vbuf[2][16 * GIN_];  // double-buffered V tiles
    async_tile(Vb, &vbuf[0][0], lane);                  // prefetch tile 0
#endif

    const int NT = C_ / 16;
    for (int it = 0; it < NT; ++it) {
        const int m0 = it * 16;

#if HAVE_ASYNC_LDS
        // Kick off DMA for the next V tile; overlaps S compute + softmax below.
        // Safe: the ds reads of buf[(it+1)&1] from iteration it-1 fed WMMAs whose
        // s_wait_dscnt retired before this wave could issue new instructions.
        if (it + 1 < NT) async_tile(Vb + (long)(m0 + 16) * GIN_, &vbuf[(it + 1) & 1][0], lane);
#else
        if (m0 + 16 < C_)
            __builtin_prefetch((const void*)(Vb + (long)(m0 + 16) * GIN_ + lane * 4), 0, 0);
#endif

        // ---- S = Q . K^T (16x16, K=32 via 8 wmma) ----
        v8f s = {};
        const float* krow = Gb + (long)(m0 + lo) * 32 + 2 * hi;  // B-layout
#pragma unroll
        for (int kt = 0; kt < 8; ++kt) {
            v2f kv; kv.x = krow[4 * kt]; kv.y = krow[4 * kt + 1];
            s = wmma4(q[kt], kv, s);
        }

        // ---- mask + relu + NEG, online softmax per row ----
        float pv[8], scale[8];
#pragma unroll
        for (int r = 0; r < 8; ++r) {
            const int og = o0 + r + 8 * hi;   // global output row of this slot
            const int mg = m0 + lo;           // global m column of this slot
            float v = s[r];
            v = (og == mg || v <= 0.0f) ? NEGV : v;
            // row max across the 16-lane group holding this row
            float t = v;
            t = fmaxf(t, __shfl_xor(t, 1, 32));
            t = fmaxf(t, __shfl_xor(t, 2, 32));
            t = fmaxf(t, __shfl_xor(t, 4, 32));
            t = fmaxf(t, __shfl_xor(t, 8, 32));
            const float mnew = fmaxf(mrun[r], t);
            scale[r] = __expf(mrun[r] - mnew);
            float p = __expf(v - mnew);
            pv[r] = p;
            float ssum = p;
            ssum += __shfl_xor(ssum, 1, 32);
            ssum += __shfl_xor(ssum, 2, 32);
            ssum += __shfl_xor(ssum, 4, 32);
            ssum += __shfl_xor(ssum, 8, 32);
            lrun[r] = lrun[r] * scale[r] + ssum;
            mrun[r] = mnew;
        }

        // ---- rescale accumulators ----
#pragma unroll
        for (int nt = 0; nt < 12; ++nt)
#pragma unroll
            for (int r = 0; r < 8; ++r) acc[nt][r] *= scale[r];

        // ---- P: D-layout -> A-layout through LDS ----
        __syncthreads();
#pragma unroll
        for (int r = 0; r < 8; ++r) pshr[(r + 8 * hi) * 16 + lo] = pv[r];
        __syncthreads();
        v2f pa[4];
#pragma unroll
        for (int kt = 0; kt < 4; ++kt) {
            pa[kt].x = pshr[lo * 16 + 4 * kt + 2 * hi];
            pa[kt].y = pshr[lo * 16 + 4 * kt + 2 * hi + 1];
        }

        // ---- wait for the current V tile, then acc += P @ V_tile ----
#if HAVE_ASYNC_LDS
        if (it + 1 < NT) __builtin_amdgcn_s_wait_asynccnt(24);  // tile it complete
        else             __builtin_amdgcn_s_wait_asynccnt(0);
        const float* vt = &vbuf[it & 1][0];   // LDS reads below (ds_load)
#else
        const float* vt = Vb + (long)m0 * GIN_;
#endif
#pragma unroll
        for (int nt = 0; nt < 12; ++nt) {
            v8f a = acc[nt];
#pragma unroll
            for (int kt = 0; kt < 4; ++kt) {
                const int kr = 4 * kt + 2 * hi;
                v2f bv;
                bv.x = vt[(long)kr * GIN_ + nt * 16 + lo];
                bv.y = vt[(long)(kr + 1) * GIN_ + nt * 16 + lo];
                a = wmma4(pa[kt], bv, a);
            }
            acc[nt] = a;
        }
    }

    // ---- finalize: divide by row sums, store AGG ----
#pragma unroll
    for (int r = 0; r < 8; ++r) {
        const float inv = 1.0f / lrun[r];
        const long rowoff = ((long)b * C_ + o0 + r + 8 * hi) * GIN_;
#pragma unroll
        for (int nt = 0; nt < 12; ++nt) AGG[rowoff + nt * 16 + lo] = acc[nt][r] * inv;
    }
}

// ---------------------------------------------------------------------------
extern "C" void kernel_launch(void* const* d_in, const int* in_sizes, int n_in,
                              void* d_out, int out_size, void* d_ws, size_t ws_size,
                              hipStream_t stream) {
    (void)in_sizes; (void)n_in; (void)out_size; (void)ws_size;

    const float* x     = (const float*)d_in[0];   // (B,C,1,T)
    const float* idemb = (const float*)d_in[1];   // (C,32)
    const float* w1    = (const float*)d_in[2];   // (128,64)
    const float* b1    = (const float*)d_in[3];
    const float* w2    = (const float*)d_in[4];   // (64,32)
    const float* b2    = (const float*)d_in[5];
    const float* ce    = (const float*)d_in[6];   // (16,32)
    const float* wic   = (const float*)d_in[7];   // (32,32)
    const float* bic   = (const float*)d_in[8];
    const float* wg    = (const float*)d_in[9];   // (64,32)
    const float* bg    = (const float*)d_in[10];
    const float* cw    = (const float*)d_in[11];  // (2048,192)
    const float* wl    = (const float*)d_in[12];  // (192,96)
    const float* bl    = (const float*)d_in[13];
    float* out = (float*)d_out;                   // (B,C,1,96)

    // workspace carve-up (floats)
    float* ws  = (float*)d_ws;
    float* H   = ws;                               // 32768*64
    float* Gm  = H   + (long)32768 * 64;           // B*C*32
    float* Vm  = Gm  + (long)B_ * C_ * 32;         // B*C*192
    float* AGG = Vm  + (long)B_ * C_ * GIN_;       // B*C*192
    float* WT  = AGG + (long)B_ * C_ * GIN_;       // B*192*192
    float* T1  = WT  + (long)B_ * GIN_ * GIN_;     // B*C*192

    const long sGE = (long)GIN_ * GIN_;            // per-batch 192x192
    const long sBC = (long)C_ * GIN_;              // per-batch 2048x192

    // 1) H = relu(X @ w1 + b1)   (32768x128 @ 128x64)
    gemm16<1, true, false><<<dim3(32768 / 16, 64 / 16, 1), 32, 0, stream>>>(
        x, 128, 1, 0,  w1, 64, 1, 0,  b1,
        nullptr, 0, 0, 0,  H, 64, 1, 0,  128);

    // 2) per-row pipeline -> G (graph_emb), V (gcn_input)
    row_pipeline<<<dim3((B_ * C_) / 256), 256, 0, stream>>>(
        x, H, idemb, w2, b2, ce, wic, bic, wg, bg, Gm, Vm);

    // 3) fused adj-softmax-aggregate (flash, async-LDS double buffered)
    flash_adj_agg<<<dim3(C_ / 16, 1, B_), 32, 0, stream>>>(Gm, Vm, AGG);

    // 4) WT[b] = tanh(AGG[b]^T @ CW)   (192x2048 @ 2048x192), A strided as transpose
    gemm16<2, false, false><<<dim3(GIN_ / 16, GIN_ / 16, B_), 32, 0, stream>>>(
        AGG, 1, GIN_, sBC,  cw, GIN_, 1, 0,  nullptr,
        nullptr, 0, 0, 0,  WT, GIN_, 1, sGE,  C_);

    // 5) T1[b] = V[b] @ WT[b] + V[b]   (2048x192 @ 192x192, residual)
    gemm16<0, false, true><<<dim3(C_ / 16, GIN_ / 16, B_), 32, 0, stream>>>(
        Vm, GIN_, 1, sBC,  WT, GIN_, 1, sGE,  nullptr,
        Vm, GIN_, 1, sBC,  T1, GIN_, 1, sBC,  GIN_);

    // 6) out = T1 @ wl + bl   (32768x192 @ 192x96)
    gemm16<0, true, false><<<dim3((B_ * C_) / 16, OUT_ / 16, 1), 32, 0, stream>>>(
        T1, GIN_, 1, 0,  wl, OUT_, 1, 0,  bl,
        nullptr, 0, 0, 0,  out, OUT_, 1, 0,  GIN_);
}